// Shear_14199161880928
// MI455X (gfx1250) — compile-verified
//
#include <hip/hip_runtime.h>

// Shear (keras apply_affine_transform, shear=50deg*frame_value, bilinear,
// mode='nearest' via coordinate clamp) on a 1x4096x4096x3 fp32 image.
//
// Bandwidth-bound gather: ~400 MB HBM traffic -> ~17us floor at 23.3 TB/s.
// No WMMA: there is no contraction dimension for the matrix pipes. Correct
// instruction mix = vmem-dominant (4x global_load_b96 gathers + b96 store +
// global_prefetch_b8), small valu, no dead libm SALU blob.

#define IMG_H 4096
#define IMG_W 4096
#define IMG_C 3

// sin/cos for theta in [0, 0.8727) rad, evaluated as fp64 Taylor then rounded
// to f32. Max abs error < 1e-8 -> below one f32 ulp of the result; avoids the
// scalarized Payne-Hanek reduction (200+ dead SALU ops) that libm sincosf
// drags in, and is more accurate than v_sin_f32 (whose error would be
// amplified by j<=4095 in src_j).
__device__ __forceinline__ void sincos_small(float xf, float* s, float* c) {
    const double x  = (double)xf;
    const double x2 = x * x;
    // sin: x * (1 + x^2*(c3 + x^2*(c5 + x^2*(c7 + x^2*c9))))   (Taylor)
    double ps = -2.505210838544171877e-8;        // -1/11! (ps is unused)
    ps =  2.755731922398589065e-6 + x2 * (-2.505210838544171877e-8 * 0.0 + ps * 0.0); // (unused guard)
    // Horner, explicit coefficients:
    const double s9 =  2.755731922398589065e-6;  //  1/9!
    const double s7 = -1.984126984126984127e-4;  // -1/7!
    const double s5 =  8.333333333333333333e-3;  //  1/5!
    const double s3 = -1.666666666666666667e-1;  // -1/3!
    double sp = s9;
    sp = s7 + x2 * sp;
    sp = s5 + x2 * sp;
    sp = s3 + x2 * sp;
    double sv = x + x * x2 * sp;

    const double c10 = -2.755731922398589065e-7; // -1/10!
    const double c8  =  2.480158730158730159e-5; //  1/8!
    const double c6  = -1.388888888888888889e-3; // -1/6!
    const double c4  =  4.166666666666666667e-2; //  1/4!
    const double c2  = -5.0e-1;                  // -1/2!
    double cp = c10;
    cp = c8 + x2 * cp;
    cp = c6 + x2 * cp;
    cp = c4 + x2 * cp;
    cp = c2 + x2 * cp;
    double cv = 1.0 + x2 * cp;

    *s = (float)sv;
    *c = (float)cv;
}

__global__ __launch_bounds__(256) void shear_bilinear_kernel(
    const float* __restrict__ img,   // [H, W, 3]
    const float* __restrict__ fvp,   // scalar frame_value
    float* __restrict__ out)         // [H, W, 3]
{
    const int j = blockIdx.x * 256 + threadIdx.x;   // column (fast axis)
    const int i = blockIdx.y;                       // row

    // Uniform scalar -> s_load; theta = deg2rad(50 * frame_value) in f32,
    // matching the jnp.float32 reference exactly up to rounding.
    const float fv    = fvp[0];
    const float theta = 50.0f * fv * 0.017453292519943295f; // * pi/180
    float sn, cs;
    sincos_small(theta, &sn, &cs);

    const float o_y = 0.5f * (float)IMG_W + 0.5f;   // 2048.5

    // src_i = clip(i - s*j + s*o_y, 0, H-1) ; src_j = clip(c*j + o_y*(1-c), 0, W-1)
    float src_i = (float)i - sn * (float)j + sn * o_y;
    src_i = fminf(fmaxf(src_i, 0.0f), (float)(IMG_H - 1));
    float src_j = cs * (float)j + o_y * (1.0f - cs);
    src_j = fminf(fmaxf(src_j, 0.0f), (float)(IMG_W - 1));

    const float i0f = floorf(src_i);
    const float j0f = floorf(src_j);
    const float fi  = src_i - i0f;
    const float fj  = src_j - j0f;

    const int i0 = (int)i0f;                  // >= 0 after clamp
    const int j0 = (int)j0f;
    const int i1 = min(i0 + 1, IMG_H - 1);
    const int j1 = min(j0 + 1, IMG_W - 1);

    const float w00 = (1.0f - fi) * (1.0f - fj);
    const float w01 = (1.0f - fi) * fj;
    const float w10 = fi * (1.0f - fj);
    const float w11 = fi * fj;

    const size_t rowElems = (size_t)IMG_W * IMG_C;
    const float* __restrict__ p00 = img + (size_t)i0 * rowElems + (size_t)j0 * IMG_C;
    const float* __restrict__ p01 = img + (size_t)i0 * rowElems + (size_t)j1 * IMG_C;
    const float* __restrict__ p10 = img + (size_t)i1 * rowElems + (size_t)j0 * IMG_C;
    const float* __restrict__ p11 = img + (size_t)i1 * rowElems + (size_t)j1 * IMG_C;

    // gfx1250 global_prefetch_b8: warm the band the next output row will
    // gather (row i0+2, same diagonal offset). Clamped -> never OOB.
    {
        const int ipf = min(i1 + 1, IMG_H - 1);
        __builtin_prefetch(img + (size_t)ipf * rowElems + (size_t)j0 * IMG_C, 0, 1);
    }

    float* __restrict__ o = out + ((size_t)i * IMG_W + j) * IMG_C;
#pragma unroll
    for (int ch = 0; ch < IMG_C; ++ch) {
        o[ch] = p00[ch] * w00 + p01[ch] * w01 + p10[ch] * w10 + p11[ch] * w11;
    }
}

extern "C" void kernel_launch(void* const* d_in, const int* in_sizes, int n_in,
                              void* d_out, int out_size, void* d_ws, size_t ws_size,
                              hipStream_t stream) {
    const float* img = (const float*)d_in[0];   // x: [1,4096,4096,3] fp32
    const float* fv  = (const float*)d_in[1];   // frame_value scalar fp32
    float* out       = (float*)d_out;           // [1,4096,4096,3] fp32

    dim3 block(256, 1, 1);                      // 8 wave32 per block
    dim3 grid(IMG_W / 256, IMG_H, 1);           // 16 x 4096 blocks
    shear_bilinear_kernel<<<grid, block, 0, stream>>>(img, fv, out);
}